// GraphHeat_9414568312942
// MI455X (gfx1250) — compile-verified
//
#include <hip/hip_runtime.h>
#include <cstdint>

#define NN   50000
#define EE   800000
#define FDIM 64
#define OUTF 32
#define KCH  10
#define BES  20

typedef float v2f __attribute__((ext_vector_type(2)));
typedef float v8f __attribute__((ext_vector_type(8)));

// ---------------- Bessel / Chebyshev coefficients (tiny) ----------------
__global__ void coeffs_kernel(const float* __restrict__ t_ptr, float* __restrict__ coeffs) {
    int k = threadIdx.x;
    if (k >= KCH) return;
    float t   = *t_ptr;
    float lt2 = logf(0.5f * t);
    float v   = (float)k;
    float iv  = 0.0f;
    for (int m = 0; m < BES; ++m) {
        float fm = (float)m;
        float lg = (2.0f * fm + v) * lt2 - lgammaf(fm + 1.0f) - lgammaf(fm + v + 1.0f);
        iv += expf(lg);
    }
    float sign = (k == 0) ? 1.0f : ((k & 1) ? -2.0f : 2.0f);
    coeffs[k] = sign * iv;
}

// ---------------- generic float4 zero ----------------
__global__ void zero4_kernel(float4* __restrict__ p, int n4) {
    int i = blockIdx.x * blockDim.x + threadIdx.x;
    if (i < n4) p[i] = make_float4(0.f, 0.f, 0.f, 0.f);
}

// ---------------- degree / normalization ----------------
__global__ void deg_kernel(const int* __restrict__ row, float* __restrict__ deg) {
    int e = blockIdx.x * blockDim.x + threadIdx.x;
    if (e < EE) unsafeAtomicAdd(&deg[row[e]], 1.0f);
}

__global__ void dinv_kernel(float* __restrict__ deg) {
    int i = blockIdx.x * blockDim.x + threadIdx.x;
    if (i < NN) {
        float d = deg[i];
        deg[i] = (d > 0.0f) ? rsqrtf(d) : 0.0f;
    }
}

__global__ void w_kernel(const int* __restrict__ row, const int* __restrict__ col,
                         const float* __restrict__ dinv, float* __restrict__ w) {
    int e = blockIdx.x * blockDim.x + threadIdx.x;
    if (e < EE) w[e] = dinv[row[e]] * dinv[col[e]];
}

// ---------------- SpMM: OUT[row] += -w * V[col]  (64 features / edge) ----------------
// 16 consecutive lanes handle one edge -> gather of V[col] is one coalesced
// 256B line (L2-resident, 12.8MB matrix), scatter is 256B of f32 atomics.
__global__ void spmm_kernel(const int* __restrict__ row, const int* __restrict__ col,
                            const float* __restrict__ w, const float* __restrict__ V,
                            float* __restrict__ OUT) {
    int gid = blockIdx.x * blockDim.x + threadIdx.x;
    int e   = gid >> 4;
    int f4  = gid & 15;
    if (e >= EE) return;
    float wv = -w[e];
    int c = col[e];
    int r = row[e];
    float4 v = ((const float4*)V)[(size_t)c * 16 + f4];
    float* o = OUT + (size_t)r * FDIM + f4 * 4;
    unsafeAtomicAdd(o + 0, wv * v.x);
    unsafeAtomicAdd(o + 1, wv * v.y);
    unsafeAtomicAdd(o + 2, wv * v.z);
    unsafeAtomicAdd(o + 3, wv * v.w);
}

// ---------------- Chebyshev combine steps ----------------
// OUT = c0*X + c1*T1
__global__ void combine_first_kernel(const float4* __restrict__ X, const float4* __restrict__ S,
                                     const float* __restrict__ coeffs, float4* __restrict__ OUT,
                                     int n4) {
    int i = blockIdx.x * blockDim.x + threadIdx.x;
    if (i >= n4) return;
    float c0 = coeffs[0], c1 = coeffs[1];
    float4 x = X[i], s = S[i];
    OUT[i] = make_float4(c0 * x.x + c1 * s.x, c0 * x.y + c1 * s.y,
                         c0 * x.z + c1 * s.z, c0 * x.w + c1 * s.w);
}

// S = 2*S - Tkm2 ;  OUT += c_k * S
__global__ void combine_step_kernel(float4* __restrict__ S, const float4* __restrict__ T2,
                                    const float* __restrict__ coeffs, int k,
                                    float4* __restrict__ OUT, int n4) {
    int i = blockIdx.x * blockDim.x + threadIdx.x;
    if (i >= n4) return;
    float ck = coeffs[k];
    float4 s = S[i], t = T2[i];
    float4 tk = make_float4(2.f * s.x - t.x, 2.f * s.y - t.y,
                            2.f * s.z - t.z, 2.f * s.w - t.w);
    S[i] = tk;
    float4 o = OUT[i];
    OUT[i] = make_float4(o.x + ck * tk.x, o.y + ck * tk.y,
                         o.z + ck * tk.z, o.w + ck * tk.w);
}

// ============================================================================
// WMMA GEMM layers.  B fragments are pre-swizzled in LDS so each lane's
// {W[ka][n], W[ka+1][n]} pair is one contiguous float2 -> single ds_load_b64
// straight into an even-aligned VGPR pair (no v_mov glue in the WMMA loop).
// Swizzled index: ((f*2 + half)*CTILES + c)*16 + m   with ka = 4*f + 2*half.
// ============================================================================

// ---------------- Layer 1: H = relu(X @ Wd + XH @ Wh)  (64 -> 64) ----------------
__global__ __launch_bounds__(256) void layer1_kernel(
    const float* __restrict__ X, const float* __restrict__ XH,
    const float* __restrict__ Wd, const float* __restrict__ Wh,
    float* __restrict__ H) {
    __shared__ v2f sWd[(FDIM / 4) * 2 * 4 * 16];   // 2048 float2 = 16KB
    __shared__ v2f sWh[(FDIM / 4) * 2 * 4 * 16];
    for (int idx = threadIdx.x; idx < 2048; idx += 256) {
        int m = idx & 15;
        int c = (idx >> 4) & 3;
        int h = (idx >> 6) & 1;
        int f = idx >> 7;
        int ka = 4 * f + 2 * h;
        int g0 = ka * FDIM + c * 16 + m;
        v2f d, e;
        d.x = Wd[g0]; d.y = Wd[g0 + FDIM];
        e.x = Wh[g0]; e.y = Wh[g0 + FDIM];
        sWd[idx] = d;
        sWh[idx] = e;
    }
    __syncthreads();

    int wave = threadIdx.x >> 5;
    int lane = threadIdx.x & 31;
    int rt   = blockIdx.x * 8 + wave;
    int rbase = rt * 16;
    if (rbase >= NN) return;            // uniform per wave: EXEC stays all-ones for WMMA
    int m    = lane & 15;
    int half = lane >> 4;
    int r = rbase + m; if (r >= NN) r = NN - 1;

    v8f acc0 = {}, acc1 = {}, acc2 = {}, acc3 = {};
    for (int kk = 0; kk < FDIM; kk += 4) {
        int f  = kk >> 2;
        int ka = kk + 2 * half;         // A 16x4 f32 layout: lanes16-31 hold K+2,K+3
        v2f ax = *(const v2f*)(X  + (size_t)r * FDIM + ka);
        v2f ah = *(const v2f*)(XH + (size_t)r * FDIM + ka);
        int bbase = ((f * 2 + half) * 4) * 16 + m;
#define TILE1(c, ACC)                                                                   \
        {                                                                               \
            v2f bd = sWd[bbase + (c) * 16];                                             \
            v2f bh = sWh[bbase + (c) * 16];                                             \
            ACC = __builtin_amdgcn_wmma_f32_16x16x4_f32(false, ax, false, bd,           \
                                                        (short)0, ACC, false, false);   \
            ACC = __builtin_amdgcn_wmma_f32_16x16x4_f32(false, ah, false, bh,           \
                                                        (short)0, ACC, false, false);   \
        }
        TILE1(0, acc0) TILE1(1, acc1) TILE1(2, acc2) TILE1(3, acc3)
#undef TILE1
    }
#pragma unroll
    for (int j = 0; j < 8; ++j) {
        int rr = rbase + j + 8 * half;  // C/D layout: VGPR j = row j (lanes0-15) / j+8 (lanes16-31)
        if (rr < NN) {
            float* hp = H + (size_t)rr * FDIM + m;
            float a0 = acc0[j]; hp[0]  = a0 > 0.f ? a0 : 0.f;
            float a1 = acc1[j]; hp[16] = a1 > 0.f ? a1 : 0.f;
            float a2 = acc2[j]; hp[32] = a2 > 0.f ? a2 : 0.f;
            float a3 = acc3[j]; hp[48] = a3 > 0.f ? a3 : 0.f;
        }
    }
}

// ---------------- Layer 2: OUT = Hd @ W1 + HH @ W2  (64 -> 32) ----------------
__global__ __launch_bounds__(256) void layer2_kernel(
    const float* __restrict__ Hd, const float* __restrict__ HH,
    const float* __restrict__ W1, const float* __restrict__ W2,
    float* __restrict__ OUT) {
    __shared__ v2f sW1[(FDIM / 4) * 2 * 2 * 16];   // 1024 float2 = 8KB
    __shared__ v2f sW2[(FDIM / 4) * 2 * 2 * 16];
    for (int idx = threadIdx.x; idx < 1024; idx += 256) {
        int m = idx & 15;
        int c = (idx >> 4) & 1;
        int h = (idx >> 5) & 1;
        int f = idx >> 6;
        int ka = 4 * f + 2 * h;
        int g0 = ka * OUTF + c * 16 + m;
        v2f d, e;
        d.x = W1[g0]; d.y = W1[g0 + OUTF];
        e.x = W2[g0]; e.y = W2[g0 + OUTF];
        sW1[idx] = d;
        sW2[idx] = e;
    }
    __syncthreads();

    int wave = threadIdx.x >> 5;
    int lane = threadIdx.x & 31;
    int rt   = blockIdx.x * 8 + wave;
    int rbase = rt * 16;
    if (rbase >= NN) return;
    int m    = lane & 15;
    int half = lane >> 4;
    int r = rbase + m; if (r >= NN) r = NN - 1;

    v8f acc0 = {}, acc1 = {};
    for (int kk = 0; kk < FDIM; kk += 4) {
        int f  = kk >> 2;
        int ka = kk + 2 * half;
        v2f a1v = *(const v2f*)(Hd + (size_t)r * FDIM + ka);
        v2f a2v = *(const v2f*)(HH + (size_t)r * FDIM + ka);
        int bbase = ((f * 2 + half) * 2) * 16 + m;
#define TILE2(c, ACC)                                                                   \
        {                                                                               \
            v2f b1 = sW1[bbase + (c) * 16];                                             \
            v2f b2 = sW2[bbase + (c) * 16];                                             \
            ACC = __builtin_amdgcn_wmma_f32_16x16x4_f32(false, a1v, false, b1,          \
                                                        (short)0, ACC, false, false);   \
            ACC = __builtin_amdgcn_wmma_f32_16x16x4_f32(false, a2v, false, b2,          \
                                                        (short)0, ACC, false, false);   \
        }
        TILE2(0, acc0) TILE2(1, acc1)
#undef TILE2
    }
#pragma unroll
    for (int j = 0; j < 8; ++j) {
        int rr = rbase + j + 8 * half;
        if (rr < NN) {
            float* op = OUT + (size_t)rr * OUTF + m;
            op[0]  = acc0[j];
            op[16] = acc1[j];
        }
    }
}

// ---------------- row log-softmax, 32 cols == wave32, in-place ----------------
__global__ void logsoftmax_kernel(float* __restrict__ out) {
    int lane = threadIdx.x & 31;
    int wave = threadIdx.x >> 5;
    int rowi = blockIdx.x * 8 + wave;
    if (rowi >= NN) return;
    float v = out[(size_t)rowi * OUTF + lane];
    float mx = v;
    for (int o = 16; o > 0; o >>= 1) mx = fmaxf(mx, __shfl_xor(mx, o, 32));
    float ex = expf(v - mx);
    float s = ex;
    for (int o = 16; o > 0; o >>= 1) s += __shfl_xor(s, o, 32);
    out[(size_t)rowi * OUTF + lane] = v - mx - logf(s);
}

// ---------------- host orchestration ----------------
extern "C" void kernel_launch(void* const* d_in, const int* in_sizes, int n_in,
                              void* d_out, int out_size, void* d_ws, size_t ws_size,
                              hipStream_t stream) {
    (void)in_sizes; (void)n_in; (void)out_size; (void)ws_size;
    const float* x    = (const float*)d_in[0];
    const int*   edge = (const int*)d_in[1];
    const float* Wd   = (const float*)d_in[2];
    const float* Wh1  = (const float*)d_in[3];
    const float* Whid = (const float*)d_in[4];
    const float* Wh2  = (const float*)d_in[5];
    const float* t    = (const float*)d_in[6];
    const int* row = edge;
    const int* col = edge + EE;

    float* ws = (float*)d_ws;
    size_t off = 0;
    float* coeffs = ws + off; off += 16;
    float* dinv   = ws + off; off += 50048;        // N padded to /16
    float* w      = ws + off; off += EE;
    const size_t NF = (size_t)NN * FDIM;
    float* S0  = ws + off; off += NF;
    float* S1  = ws + off; off += NF;
    float* S2  = ws + off; off += NF;
    float* xh  = ws + off; off += NF;
    float* hid = ws + off; off += NF;
    float* hh  = ws + off; off += NF;

    const int B = 256;
    const int n4 = (int)(NF / 4);                  // 800000
    const int zg = (n4 + B - 1) / B;
    const int sg = (EE * 16 + B - 1) / B;
    const int eg = (EE + B - 1) / B;
    const int tiles = (NN + 15) / 16;              // 3125
    const int gemmg = (tiles + 7) / 8;             // 391

    coeffs_kernel<<<1, 32, 0, stream>>>(t, coeffs);
    zero4_kernel<<<(50048 / 4 + B - 1) / B, B, 0, stream>>>((float4*)dinv, 50048 / 4);
    deg_kernel<<<eg, B, 0, stream>>>(row, dinv);
    dinv_kernel<<<(NN + B - 1) / B, B, 0, stream>>>(dinv);
    w_kernel<<<eg, B, 0, stream>>>(row, col, dinv, w);

    auto heat = [&](const float* X, float* OUT) {
        zero4_kernel<<<zg, B, 0, stream>>>((float4*)S0, n4);
        spmm_kernel<<<sg, B, 0, stream>>>(row, col, w, X, S0);
        combine_first_kernel<<<zg, B, 0, stream>>>((const float4*)X, (const float4*)S0,
                                                   coeffs, (float4*)OUT, n4);
        const float* Tkm2 = X;
        float* Tkm1 = S0;
        float* bufs[3] = {S0, S1, S2};
        int nxt = 1;
        for (int k = 2; k < KCH; ++k) {
            float* Snew = bufs[nxt];
            zero4_kernel<<<zg, B, 0, stream>>>((float4*)Snew, n4);
            spmm_kernel<<<sg, B, 0, stream>>>(row, col, w, Tkm1, Snew);
            combine_step_kernel<<<zg, B, 0, stream>>>((float4*)Snew, (const float4*)Tkm2,
                                                      coeffs, k, (float4*)OUT, n4);
            Tkm2 = Tkm1;
            Tkm1 = Snew;
            nxt = (nxt + 1) % 3;
        }
    };

    heat(x, xh);
    layer1_kernel<<<gemmg, 256, 0, stream>>>(x, xh, Wd, Wh1, hid);
    heat(hid, hh);
    layer2_kernel<<<gemmg, 256, 0, stream>>>(hid, hh, Whid, Wh2, (float*)d_out);
    logsoftmax_kernel<<<(NN + 7) / 8, 256, 0, stream>>>((float*)d_out);
}